// SGC_P_1623497638172
// MI455X (gfx1250) — compile-verified
//
#include <hip/hip_runtime.h>
#include <hip/hip_bf16.h>

typedef float v2f __attribute__((ext_vector_type(2)));
typedef float v8f __attribute__((ext_vector_type(8)));

#define F_IN 32
#define F_OUT 64

// ---------------------------------------------------------------------------
// deg[i] = 1.0 (self-loop)
__global__ void sgc_init_deg(float* __restrict__ deg, int N) {
    int i = blockIdx.x * blockDim.x + threadIdx.x;
    if (i < N) deg[i] = 1.0f;
}

// deg[dst[e]] += 1
__global__ void sgc_count_deg(const int* __restrict__ dst, float* __restrict__ deg, long E) {
    long e = (long)blockIdx.x * blockDim.x + threadIdx.x;
    if (e < E) atomicAdd(&deg[dst[e]], 1.0f);
}

// dinv[i] = rsqrt(deg[i])   (deg >= 1 always, self-loop)
__global__ void sgc_dinv(float* __restrict__ deg, int N) {
    int i = blockIdx.x * blockDim.x + threadIdx.x;
    if (i < N) deg[i] = __frsqrt_rn(deg[i]);
}

// Self-loop contribution initializes the accumulator:
// h_next[i][f] = dinv[i]^2 * h_cur[i][f]
__global__ void sgc_selfloop(const float* __restrict__ dinv,
                             const float* __restrict__ hcur,
                             float* __restrict__ hnext, int N) {
    long t = (long)blockIdx.x * blockDim.x + threadIdx.x;
    int i = (int)(t >> 5);
    int f = (int)(t & 31);
    if (i >= N) return;
    float di = dinv[i];
    hnext[(long)i * F_IN + f] = di * di * hcur[(long)i * F_IN + f];
}

// Edge scatter: one wave32 per edge, lane = feature -> fully coalesced
// gather of h[src] and coalesced f32 atomic adds into h[dst] (L2-resident:
// each 12.8 MB h buffer and the edge lists all fit in the 192 MB L2).
__global__ void sgc_scatter(const int* __restrict__ src,
                            const int* __restrict__ dst,
                            const float* __restrict__ dinv,
                            const float* __restrict__ hcur,
                            float* __restrict__ hnext, long E) {
    long t = (long)blockIdx.x * blockDim.x + threadIdx.x;
    long e = t >> 5;
    int f = (int)(t & 31);
    if (e >= E) return;
    int s = src[e];
    int d = dst[e];
    float norm = dinv[s] * dinv[d];
    atomicAdd(&hnext[(long)d * F_IN + f], norm * hcur[(long)s * F_IN + f]);
}

// ---------------------------------------------------------------------------
// Dense finish: out[N,64] = h[N,32] @ W^T + b using V_WMMA_F32_16X16X4_F32.
// One wave computes one 16(M) x 16(N) tile of out, looping K=32 in steps of 4.
//
// A (16x4 f32): lane M = lane&15; lanes 0-15 hold K={k0,k0+1}, lanes 16-31
//               hold K={k0+2,k0+3} in VGPR0/VGPR1.
// B (4x16 f32): mirrored — lane N = lane&15, same K split per lane half.
//               B[k][n] = W[n][k]  (W stored [64,32] row-major).
// C/D (16x16 f32, 8 VGPRs): VGPR r, lanes 0-15 -> M=r, lanes 16-31 -> M=8+r.
//
// Bias trick: every C VGPR has the same output column N per lane, so the
// accumulator is seeded with bias[col] instead of adding it afterwards.
__global__ __launch_bounds__(256)
void sgc_gemm_wmma(const float* __restrict__ h,
                   const float* __restrict__ W,
                   const float* __restrict__ bias,
                   float* __restrict__ out, int N) {
    const int wave   = (blockIdx.x * blockDim.x + threadIdx.x) >> 5;
    const int lane   = threadIdx.x & 31;
    const int tilesM = (N + 15) >> 4;
    const int tilesN = F_OUT / 16;            // 4
    if (wave >= tilesM * tilesN) return;      // uniform per wave
    const int tileM   = wave / tilesN;
    const int tileN   = wave - tileM * tilesN;
    const int rowBase = tileM * 16;
    const int colBase = tileN * 16;
    const int half    = lane >> 4;            // 0 | 1
    const int l15     = lane & 15;

    // clamp row (not predicate) so EXEC stays all-ones for WMMA
    int arow = rowBase + l15;
    if (arow >= N) arow = N - 1;
    const float* hrow = h + (long)arow * F_IN;
    const float* wrow = W + (long)(colBase + l15) * F_IN;

    // seed accumulator with bias (same output column for all 8 C VGPRs)
    const float bv = bias[colBase + l15];
    v8f c;
#pragma unroll
    for (int r = 0; r < 8; ++r) c[r] = bv;

#pragma unroll
    for (int k0 = 0; k0 < F_IN; k0 += 4) {
        const int ka = k0 + half * 2;
        v2f a, b;
        a.x = hrow[ka];
        a.y = hrow[ka + 1];
        b.x = wrow[ka];
        b.y = wrow[ka + 1];
        c = __builtin_amdgcn_wmma_f32_16x16x4_f32(
                /*neg_a=*/false, a, /*neg_b=*/false, b,
                /*c_mod=*/(short)0, c, /*reuse_a=*/false, /*reuse_b=*/false);
    }

    float* o = out + (long)(rowBase + half * 8) * F_OUT + colBase + l15;
    if (rowBase + 16 <= N) {
        // full tile (always taken when N % 16 == 0): unconditional stores
#pragma unroll
        for (int r = 0; r < 8; ++r) o[(long)r * F_OUT] = c[r];
    } else {
        // ragged last tile: predicated stores (after all WMMAs -> legal)
#pragma unroll
        for (int r = 0; r < 8; ++r) {
            if (rowBase + half * 8 + r < N) o[(long)r * F_OUT] = c[r];
        }
    }
}

// ---------------------------------------------------------------------------
extern "C" void kernel_launch(void* const* d_in, const int* in_sizes, int n_in,
                              void* d_out, int out_size, void* d_ws, size_t ws_size,
                              hipStream_t stream) {
    const float* x    = (const float*)d_in[0];   // [N,32]
    const int*   ei   = (const int*)d_in[1];     // [2,E] row-major
    const float* W    = (const float*)d_in[2];   // [64,32]
    const float* bias = (const float*)d_in[3];   // [64]
    float*       out  = (float*)d_out;           // [N,64]

    const int  N = in_sizes[0] / F_IN;
    const long E = (long)in_sizes[1] / 2;
    const int* src = ei;
    const int* dst = ei + E;

    // workspace: dinv (N, padded), hA (N*32), hB (N*32)
    float* deg = (float*)d_ws;
    size_t npad = ((size_t)N + 255) & ~(size_t)255;
    float* hA = deg + npad;
    float* hB = hA + (size_t)N * F_IN;

    const int B = 256;

    // 1) degree + dinv
    sgc_init_deg<<<(N + B - 1) / B, B, 0, stream>>>(deg, N);
    sgc_count_deg<<<(int)((E + B - 1) / B), B, 0, stream>>>(dst, deg, E);
    sgc_dinv<<<(N + B - 1) / B, B, 0, stream>>>(deg, N);

    const long nodeThreads = (long)N * F_IN;
    const long edgeThreads = E * F_IN;
    const int  nodeBlocks  = (int)((nodeThreads + B - 1) / B);
    const int  edgeBlocks  = (int)((edgeThreads + B - 1) / B);

    // 2) K = 3 propagation hops (ping-pong hA/hB; hop 1 reads x directly)
    sgc_selfloop<<<nodeBlocks, B, 0, stream>>>(deg, x, hA, N);
    sgc_scatter <<<edgeBlocks, B, 0, stream>>>(src, dst, deg, x, hA, E);

    sgc_selfloop<<<nodeBlocks, B, 0, stream>>>(deg, hA, hB, N);
    sgc_scatter <<<edgeBlocks, B, 0, stream>>>(src, dst, deg, hA, hB, E);

    sgc_selfloop<<<nodeBlocks, B, 0, stream>>>(deg, hB, hA, N);
    sgc_scatter <<<edgeBlocks, B, 0, stream>>>(src, dst, deg, hB, hA, E);

    // 3) dense finish via WMMA: out = hA @ W^T + b (bias pre-seeded in C)
    const int tiles   = ((N + 15) / 16) * (F_OUT / 16);
    const int threads = tiles * 32;
    sgc_gemm_wmma<<<(threads + 255) / 256, 256, 0, stream>>>(hA, W, bias, out, N);
}